// GNNRecommender_87514253623723
// MI455X (gfx1250) — compile-verified
//
#include <hip/hip_runtime.h>

typedef __attribute__((ext_vector_type(2))) float v2f;
typedef __attribute__((ext_vector_type(8))) float v8f;

#define FEAT 64

// ---------------- degree / norm kernels ----------------

__global__ __launch_bounds__(256) void k_deg_init(float* deg, int n) {
  int i = blockIdx.x * blockDim.x + threadIdx.x;
  if (i < n) deg[i] = 1.0f;  // self-loop contributes 1 to every node's degree
}

__global__ __launch_bounds__(256) void k_deg_accum(const long long* __restrict__ dst,
                                                   float* deg, int E) {
  int e = blockIdx.x * blockDim.x + threadIdx.x;
  if (e < E) atomicAdd(&deg[(int)dst[e]], 1.0f);
}

__global__ __launch_bounds__(256) void k_rsqrt_inplace(float* deg, int n) {
  int i = blockIdx.x * blockDim.x + threadIdx.x;
  if (i < n) deg[i] = rsqrtf(deg[i]);  // deg >= 1 always (self-loop), no zero guard needed
}

// ---------------- fused GEMM (WMMA f32 16x16x4) + self-loop/bias epilogue ----------------
// Computes XW = act(X) @ W   (act = relu if RELU, compile-time)
// and      OutInit = bias + XW * dinv^2   (the self-loop term of the scatter-sum)
// One wave per 16-row tile; exact f32 math via V_WMMA_F32_16X16X4_F32.
// NOTE: X and OutInit may alias (layer 2): each wave reads only its own 16 rows
// of X before writing those same rows of OutInit -> tile-local RAW only, safe.

template <bool RELU>
__global__ __launch_bounds__(256)
void k_gemm_wmma(const float* X, const float* __restrict__ W,
                 const float* __restrict__ bias, const float* __restrict__ dinv,
                 float* __restrict__ XW, float* OutInit, int n_rows)
{
  int wave = (int)((blockIdx.x * blockDim.x + threadIdx.x) >> 5);
  int lane = (int)(threadIdx.x & 31);
  int row_base = wave * 16;
  if (row_base >= n_rows) return;

  int half = lane >> 4;   // 0: lanes 0-15, 1: lanes 16-31
  int l16  = lane & 15;

  // ---- A-matrix preload: 16x4 f32 per WMMA, 16 K-steps covering K=0..63 ----
  // ISA layout (32-bit A 16x4): lanes 0-15 hold M=0..15 with VGPR0=K0,VGPR1=K1;
  // lanes 16-31 hold M=0..15 with VGPR0=K2,VGPR1=K3.
  int arow = row_base + l16;
  if (arow >= n_rows) arow = n_rows - 1;  // clamp (N is a multiple of 16 here anyway)
  const float* xrow = X + (size_t)arow * FEAT;

  v2f A[16];
#pragma unroll
  for (int kt = 0; kt < 16; ++kt) {
    float a0 = xrow[4 * kt + 2 * half + 0];
    float a1 = xrow[4 * kt + 2 * half + 1];
    if (RELU) { a0 = fmaxf(a0, 0.0f); a1 = fmaxf(a1, 0.0f); }
    A[kt].x = a0;
    A[kt].y = a1;
  }

#pragma unroll
  for (int nt = 0; nt < 4; ++nt) {
    int col = nt * 16 + l16;
    v8f c = {};
#pragma unroll
    for (int kt = 0; kt < 16; ++kt) {
      // B-matrix 4x16 f32: lanes 0-15 N=0..15 with VGPR0=K0,VGPR1=K1;
      // lanes 16-31 N=0..15 with VGPR0=K2,VGPR1=K3. W is row-major [K][N].
      v2f b;
      b.x = W[(4 * kt + 2 * half + 0) * FEAT + col];
      b.y = W[(4 * kt + 2 * half + 1) * FEAT + col];
      c = __builtin_amdgcn_wmma_f32_16x16x4_f32(
              /*neg_a=*/false, A[kt], /*neg_b=*/false, b,
              /*c_mod=*/(short)0, c, /*reuse_a=*/false, /*reuse_b=*/false);
    }
    // ---- epilogue: D layout = VGPR r -> (M = r + 8*half, N = l16) ----
    float bcol = bias[col];
#pragma unroll
    for (int r = 0; r < 8; ++r) {
      int row = row_base + r + 8 * half;
      if (row < n_rows) {
        float v = c[r];
        XW[(size_t)row * FEAT + col] = v;
        float di = dinv[row];
        OutInit[(size_t)row * FEAT + col] = fmaf(v, di * di, bcol);
      }
    }
  }
}

// ---------------- edge gather/scale/scatter-add ----------------
// One wave32 per edge: lane l covers features l and l+32.
// xw rows (25.6 MB) and out rows (25.6 MB) are L2-resident on MI455X (192 MB L2),
// so both the random gather and the f32 atomics resolve in L2.

__global__ __launch_bounds__(256)
void k_edge_scatter(const long long* __restrict__ src, const long long* __restrict__ dst,
                    const float* __restrict__ dinv, const float* __restrict__ XW,
                    float* Out, int E)
{
  int wave = (int)(blockIdx.x * 8 + (threadIdx.x >> 5));
  int lane = (int)(threadIdx.x & 31);
  if (wave >= E) return;
  int s = (int)src[wave];
  int d = (int)dst[wave];
  float norm = dinv[s] * dinv[d];
  const float* xs = XW + (size_t)s * FEAT;
  float* od = Out + (size_t)d * FEAT;
  atomicAdd(&od[lane],      xs[lane]      * norm);
  atomicAdd(&od[lane + 32], xs[lane + 32] * norm);
}

// ---------------- launcher ----------------

extern "C" void kernel_launch(void* const* d_in, const int* in_sizes, int n_in,
                              void* d_out, int out_size, void* d_ws, size_t ws_size,
                              hipStream_t stream)
{
  const float*     x    = (const float*)d_in[0];
  const long long* edge = (const long long*)d_in[1];   // int64 [2, E]
  const float*     W1   = (const float*)d_in[2];
  const float*     b1   = (const float*)d_in[3];
  const float*     W2   = (const float*)d_in[4];
  const float*     b2   = (const float*)d_in[5];

  int N = in_sizes[0] / FEAT;
  int E = in_sizes[1] / 2;
  const long long* src = edge;       // edge_index[0]
  const long long* dst = edge + E;   // edge_index[1]

  // workspace layout: [ dinv: N floats | XW: N*64 floats ]  (~26 MB total)
  float* dinv = (float*)d_ws;
  size_t off  = (((size_t)N * sizeof(float)) + 255) & ~(size_t)255;
  float* XW   = (float*)((char*)d_ws + off);
  float* out  = (float*)d_out;

  const int T = 256;          // 8 wave32 per block
  const int WPB = T / 32;

  // normalization: deg = 1 + indeg;  dinv = rsqrt(deg)
  k_deg_init  <<<(N + T - 1) / T, T, 0, stream>>>(dinv, N);
  k_deg_accum <<<(E + T - 1) / T, T, 0, stream>>>(dst, dinv, E);
  k_rsqrt_inplace<<<(N + T - 1) / T, T, 0, stream>>>(dinv, N);

  int row_tiles  = (N + 15) / 16;
  int gemm_grid  = (row_tiles + WPB - 1) / WPB;
  int edge_grid  = (E + WPB - 1) / WPB;

  // Layer 1: XW = x@W1 ; out = b1 + XW*dinv^2 (self-loop) ; scatter edges
  k_gemm_wmma<false><<<gemm_grid, T, 0, stream>>>(x, W1, b1, dinv, XW, out, N);
  k_edge_scatter    <<<edge_grid, T, 0, stream>>>(src, dst, dinv, XW, out, E);

  // Layer 2: XW = relu(out)@W2 ; out = b2 + XW*dinv^2 ; scatter edges
  k_gemm_wmma<true> <<<gemm_grid, T, 0, stream>>>(out, W2, b2, dinv, XW, out, N);
  k_edge_scatter    <<<edge_grid, T, 0, stream>>>(src, dst, dinv, XW, out, E);
}